// GraphConvolutionBS_1967095022032
// MI455X (gfx1250) — compile-verified
//
#include <hip/hip_runtime.h>
#include <hip/hip_bf16.h>

// GCN layer for MI455X (gfx1250, wave32).
// [K1] support = X@W (fp32 WMMA) -> [K2] edge scatter (L2 atomics)
// -> [K3] h = accum + X@Wself + bias + fused BN stats (fp32 WMMA)
// -> [K4] BN finalize -> [K5] out = relu(h*sc+sh) + x
//
// W is staged in LDS *transposed* with a 132-float row stride:
//  - each lane's B fragment (K,K+1 at fixed n) is one ds_load_b64 into an
//    even-aligned pair (no repack movs),
//  - banks (4n+k+khi) mod 64: lanes 0-15 and 16-31 tile disjoint bank pairs.
// Each wave computes a 32-row x 128-col strip: two A fragments reuse every
// B fragment -> 4x fewer DS ops per WMMA than round 1.

#define NODES_D 128
#define LDS_STRIDE 132                       // floats; bank-conflict-free pad
#define LDS_BYTES (NODES_D * LDS_STRIDE * 4) // 67584 B (dynamic LDS, 320KB/WGP)
#define BN_EPS 1e-5f

typedef __attribute__((ext_vector_type(2))) float v2f;
typedef __attribute__((ext_vector_type(8))) float v8f;

__device__ __forceinline__ v8f wmma_f32(v2f a, v2f b, v8f c) {
  return __builtin_amdgcn_wmma_f32_16x16x4_f32(false, a, false, b, (short)0, c,
                                               false, false);
}

// Cooperative transposed load of W[128][128] into lWt[n*LDS_STRIDE + k].
__device__ __forceinline__ void load_w_transposed(float* lWt,
                                                  const float* __restrict__ W,
                                                  int tid) {
  for (int i = tid; i < NODES_D * (NODES_D / 4); i += 128) {
    const int kk = i >> 5;          // source row (k)
    const int j  = i & 31;          // float4 chunk along n
    const float4 w = ((const float4*)(W + (size_t)kk * NODES_D))[j];
    const int n0 = j << 2;
    lWt[(n0 + 0) * LDS_STRIDE + kk] = w.x;
    lWt[(n0 + 1) * LDS_STRIDE + kk] = w.y;
    lWt[(n0 + 2) * LDS_STRIDE + kk] = w.z;
    lWt[(n0 + 3) * LDS_STRIDE + kk] = w.w;
  }
}

// ---------------------------------------------------------------------------
// K1: support = X @ W.  One wave per 32-row strip.
// A frag (16x4 f32): lanes 0-15 hold K0,K1; lanes 16-31 hold K2,K3 (same M set).
// B frag (4x16 f32): lanes 0-15 N=lane K0,K1; lanes 16-31 same N, K2,K3.
// C/D (16x16 f32): lanes 0-15 -> M=0..7 in c[0..7]; lanes 16-31 -> M=8..15.
// ---------------------------------------------------------------------------
__global__ __launch_bounds__(128) void gemm_support_kernel(
    const float* __restrict__ X, const float* __restrict__ W,
    float* __restrict__ Y, int N) {
  extern __shared__ float lWt[];
  const int tid = threadIdx.x;
  load_w_transposed(lWt, W, tid);
  __syncthreads();

  const int lane = tid & 31;
  const int mlo  = lane & 15;
  const int khi  = (lane >> 4) << 1;           // 0 | 2
  const int wid  = (blockIdx.x * blockDim.x + tid) >> 5;
  const int nw   = (gridDim.x * blockDim.x) >> 5;
  const int nStrips = (N + 31) >> 5;
  const float* lbase = lWt + mlo * LDS_STRIDE + khi;  // +t*16*LDS_STRIDE+k imm

  for (int s = wid; s < nStrips; s += nw) {
    const int row0 = s << 5;
    int ar0 = row0 + mlo;       if (ar0 > N - 1) ar0 = N - 1;
    int ar1 = row0 + 16 + mlo;  if (ar1 > N - 1) ar1 = N - 1;
    const float* A0 = X + (size_t)ar0 * NODES_D + khi;
    const float* A1 = X + (size_t)ar1 * NODES_D + khi;

    v8f acc[16] = {};
#pragma unroll 2
    for (int k = 0; k < NODES_D; k += 4) {
      const v2f a0 = *(const v2f*)(A0 + k);
      const v2f a1 = *(const v2f*)(A1 + k);
#pragma unroll
      for (int t = 0; t < 8; ++t) {
        const v2f b = *(const v2f*)(lbase + t * 16 * LDS_STRIDE + k);
        acc[t]     = wmma_f32(a0, b, acc[t]);
        acc[8 + t] = wmma_f32(a1, b, acc[8 + t]);
      }
    }

#pragma unroll
    for (int h = 0; h < 2; ++h) {
      const int rbase = row0 + (h << 4) + ((lane < 16) ? 0 : 8);
#pragma unroll
      for (int t = 0; t < 8; ++t) {
        const int n = (t << 4) + mlo;
#pragma unroll
        for (int i = 0; i < 8; ++i) {
          const int r = rbase + i;
          if (r < N) Y[(size_t)r * NODES_D + n] = acc[(h << 3) + t][i];
        }
      }
    }
  }
}

// ---------------------------------------------------------------------------
// K2: accum[row[e]] += val[e] * support[col[e]]  (one wave/edge, float4/lane)
// ---------------------------------------------------------------------------
__global__ __launch_bounds__(256) void edge_scatter_kernel(
    const float* __restrict__ support, const int* __restrict__ erow,
    const int* __restrict__ ecol, const float* __restrict__ eval,
    float* __restrict__ accum, int E) {
  const int lane = threadIdx.x & 31;
  const int warp = (blockIdx.x * blockDim.x + threadIdx.x) >> 5;
  const int nwarps = (gridDim.x * blockDim.x) >> 5;

  for (int e = warp; e < E; e += nwarps) {
    if (e + nwarps < E && lane == 0) {        // global_prefetch_b8 streaming
      __builtin_prefetch(&ecol[e + nwarps], 0, 1);
      __builtin_prefetch(&erow[e + nwarps], 0, 1);
      __builtin_prefetch(&eval[e + nwarps], 0, 1);
    }
    const int r = erow[e];
    const int c = ecol[e];
    const float v = eval[e];
    const float4 s = ((const float4*)(support + (size_t)c * NODES_D))[lane];
    float* dst = accum + (size_t)r * NODES_D + lane * 4;
    atomicAdd(dst + 0, v * s.x);
    atomicAdd(dst + 1, v * s.y);
    atomicAdd(dst + 2, v * s.z);
    atomicAdd(dst + 3, v * s.w);
  }
}

// ---------------------------------------------------------------------------
// K3: h = accum + X @ Wself + bias; fused per-feature sum/sumsq for BatchNorm.
// ---------------------------------------------------------------------------
__global__ __launch_bounds__(128) void gemm_self_bn_kernel(
    const float* __restrict__ X, const float* __restrict__ Wself,
    const float* __restrict__ bias, const float* __restrict__ accum,
    float* __restrict__ H, float* __restrict__ stats, int N) {
  extern __shared__ float lWt[];
  const int tid = threadIdx.x;
  load_w_transposed(lWt, Wself, tid);
  __syncthreads();

  const int lane = tid & 31;
  const int mlo  = lane & 15;
  const int khi  = (lane >> 4) << 1;
  const int wid  = (blockIdx.x * blockDim.x + tid) >> 5;
  const int nw   = (gridDim.x * blockDim.x) >> 5;
  const int nStrips = (N + 31) >> 5;
  const float* lbase = lWt + mlo * LDS_STRIDE + khi;

  float bn_[8];
  float ssum[8] = {0.f, 0.f, 0.f, 0.f, 0.f, 0.f, 0.f, 0.f};
  float ssq[8]  = {0.f, 0.f, 0.f, 0.f, 0.f, 0.f, 0.f, 0.f};
#pragma unroll
  for (int t = 0; t < 8; ++t) bn_[t] = bias[(t << 4) + mlo];

  for (int s = wid; s < nStrips; s += nw) {
    const int row0 = s << 5;
    int ar0 = row0 + mlo;       if (ar0 > N - 1) ar0 = N - 1;
    int ar1 = row0 + 16 + mlo;  if (ar1 > N - 1) ar1 = N - 1;
    const float* A0 = X + (size_t)ar0 * NODES_D + khi;
    const float* A1 = X + (size_t)ar1 * NODES_D + khi;

    v8f acc[16] = {};
#pragma unroll 2
    for (int k = 0; k < NODES_D; k += 4) {
      const v2f a0 = *(const v2f*)(A0 + k);
      const v2f a1 = *(const v2f*)(A1 + k);
#pragma unroll
      for (int t = 0; t < 8; ++t) {
        const v2f b = *(const v2f*)(lbase + t * 16 * LDS_STRIDE + k);
        acc[t]     = wmma_f32(a0, b, acc[t]);
        acc[8 + t] = wmma_f32(a1, b, acc[8 + t]);
      }
    }

#pragma unroll
    for (int h = 0; h < 2; ++h) {
      const int rbase = row0 + (h << 4) + ((lane < 16) ? 0 : 8);
#pragma unroll
      for (int t = 0; t < 8; ++t) {
        const int n = (t << 4) + mlo;
#pragma unroll
        for (int i = 0; i < 8; ++i) {
          const int r = rbase + i;
          if (r < N) {
            const float v =
                acc[(h << 3) + t][i] + accum[(size_t)r * NODES_D + n] + bn_[t];
            H[(size_t)r * NODES_D + n] = v;
            ssum[t] += v;
            ssq[t]  += v * v;
          }
        }
      }
    }
  }

  // lanes L and L+16 own the same feature column -> fold, then one atomic each
#pragma unroll
  for (int t = 0; t < 8; ++t) {
    const int n = (t << 4) + mlo;
    const float s1 = ssum[t] + __shfl_xor(ssum[t], 16, 32);
    const float s2 = ssq[t]  + __shfl_xor(ssq[t], 16, 32);
    if (lane < 16) {
      atomicAdd(&stats[n], s1);
      atomicAdd(&stats[NODES_D + n], s2);
    }
  }
}

// ---------------------------------------------------------------------------
// K4: per-feature scale/shift from batch stats (biased variance)
// ---------------------------------------------------------------------------
__global__ __launch_bounds__(128) void bn_finalize_kernel(
    const float* __restrict__ stats, const float* __restrict__ gamma,
    const float* __restrict__ beta, float* __restrict__ sc, int N) {
  const int n = threadIdx.x;
  const float invN = 1.0f / (float)N;
  const float mean = stats[n] * invN;
  const float var  = stats[NODES_D + n] * invN - mean * mean;
  const float s = gamma[n] * rsqrtf(var + BN_EPS);
  sc[n] = s;
  sc[NODES_D + n] = beta[n] - mean * s;
}

// ---------------------------------------------------------------------------
// K5: out = relu(h * scale + shift) + x   (float4 streaming)
// ---------------------------------------------------------------------------
__global__ __launch_bounds__(256) void apply_kernel(
    const float* __restrict__ H, const float* __restrict__ X,
    const float* __restrict__ sc, float* __restrict__ out, int total4) {
  const int stride = gridDim.x * blockDim.x;
  for (int i = blockIdx.x * blockDim.x + threadIdx.x; i < total4; i += stride) {
    const int n0 = (i << 2) & (NODES_D - 1);
    const float4 hv = ((const float4*)H)[i];
    const float4 xv = ((const float4*)X)[i];
    const float4 s  = *(const float4*)(sc + n0);
    const float4 b  = *(const float4*)(sc + NODES_D + n0);
    float4 o;
    o.x = fmaxf(fmaf(hv.x, s.x, b.x), 0.f) + xv.x;
    o.y = fmaxf(fmaf(hv.y, s.y, b.y), 0.f) + xv.y;
    o.z = fmaxf(fmaf(hv.z, s.z, b.z), 0.f) + xv.z;
    o.w = fmaxf(fmaf(hv.w, s.w, b.w), 0.f) + xv.w;
    ((float4*)out)[i] = o;
  }
}

// ---------------------------------------------------------------------------
extern "C" void kernel_launch(void* const* d_in, const int* in_sizes, int n_in,
                              void* d_out, int out_size, void* d_ws, size_t ws_size,
                              hipStream_t stream) {
  const float* x      = (const float*)d_in[0];
  const int*   erow   = (const int*)d_in[1];
  const int*   ecol   = (const int*)d_in[2];
  const float* eval   = (const float*)d_in[3];
  const float* weight = (const float*)d_in[4];
  const float* wself  = (const float*)d_in[5];
  const float* bias   = (const float*)d_in[6];
  const float* gamma  = (const float*)d_in[7];
  const float* beta   = (const float*)d_in[8];
  float* out = (float*)d_out;

  const int N = in_sizes[0] / NODES_D;   // 100000
  const int E = in_sizes[1];             // 1600000
  const size_t nd = (size_t)N * NODES_D;

  float* support = (float*)d_ws;            // N*D   (later reused as H)
  float* accum   = support + nd;            // N*D   scatter target
  float* stats   = accum + nd;              // 256 floats (sum | sumsq)
  float* sc      = stats + 2 * NODES_D;     // 256 floats (scale | shift)

  hipMemsetAsync(accum, 0, nd * sizeof(float), stream);
  hipMemsetAsync(stats, 0, 2 * NODES_D * sizeof(float), stream);

  const int nStrips = (N + 31) >> 5;              // 3125 strips of 32 rows
  const int gemmBlocks = (nStrips + 3) / 4;       // 4 waves/block

  gemm_support_kernel<<<gemmBlocks, 128, LDS_BYTES, stream>>>(x, weight,
                                                              support, N);
  edge_scatter_kernel<<<4096, 256, 0, stream>>>(support, erow, ecol, eval,
                                                accum, E);
  gemm_self_bn_kernel<<<gemmBlocks, 128, LDS_BYTES, stream>>>(
      x, wself, bias, accum, support, stats, N);
  bn_finalize_kernel<<<1, 128, 0, stream>>>(stats, gamma, beta, sc, N);
  apply_kernel<<<4096, 256, 0, stream>>>(support, x, sc, out, (int)(nd / 4));
}